// SubnetworkEncoder_74285754351672
// MI455X (gfx1250) — compile-verified
//
#include <hip/hip_runtime.h>

// ---------------------------------------------------------------------------
// Hetero-GCN (5 node types, 8 relations, D=128) for MI455X / gfx1250.
//  - fp32 WMMA (V_WMMA_F32_16X16X4_F32) for the 8 feature GEMMs
//  - L2-resident gather + native f32 global atomics for the edge scatter
//  - dst-norm folded into edge contributions; GraphConv bias skipped
//    (it cancels exactly through BatchNorm's mean subtraction)
// ---------------------------------------------------------------------------

#define D_FEAT 128

typedef __attribute__((ext_vector_type(2))) float v2f;
typedef __attribute__((ext_vector_type(8))) float v8f;

__device__ __forceinline__ void atomAddF32(float* p, float v) {
  // Lowers to global_atomic_add_f32 (no CAS loop) on gfx1250.
  unsafeAtomicAdd(p, v);
}

// -------------------- degree counting + norm finalize ----------------------

__global__ __launch_bounds__(256)
void count_deg_kernel(const int* __restrict__ src, const int* __restrict__ dst,
                      int* __restrict__ cs, int* __restrict__ cd, int nE) {
  int i = blockIdx.x * 256 + threadIdx.x;
  if (i >= nE) return;
  atomicAdd(&cs[src[i]], 1);
  atomicAdd(&cd[dst[i]], 1);
}

__global__ __launch_bounds__(256)
void finalize_norm_kernel(void* __restrict__ buf, int n) {
  int i = blockIdx.x * 256 + threadIdx.x;
  if (i >= n) return;
  int c = ((const int*)buf)[i];
  ((float*)buf)[i] = (c > 0) ? rsqrtf((float)c) : 0.0f;
}

// -------------------- fp32 WMMA GEMM:  h = (x * ns[row]) @ W ---------------
// One wave (32 lanes) computes a 16-row stripe across all 128 output columns:
// 8 accumulator tiles of 16x16, K swept 4 at a time with V_WMMA_F32_16X16X4_F32.
// A-frag layout (ISA 16x4 f32): lanes 0-15 -> M=lane, K=k0,k0+1 in v[0],v[1];
//                               lanes 16-31 -> M=lane-16, K=k0+2,k0+3.
// C/D layout: VGPR j: lanes0-15 = (M=j, N=lane), lanes16-31 = (M=j+8, N=lane-16).

__global__ __launch_bounds__(32)
void gemm_ns_wmma_kernel(const float* __restrict__ x, const float* __restrict__ nsv,
                         const float* __restrict__ W, float* __restrict__ h) {
  const int lane = threadIdx.x;
  const int half = lane >> 4;     // 0: K low pair, 1: K high pair
  const int l16  = lane & 15;
  const int row0 = blockIdx.x << 4;
  const int m    = row0 + l16;
  const float s  = nsv[m];
  const float* __restrict__ xrow = x + (size_t)m * D_FEAT;

  v8f acc[8];
  const v8f vzero = {0.f, 0.f, 0.f, 0.f, 0.f, 0.f, 0.f, 0.f};
#pragma unroll
  for (int t = 0; t < 8; ++t) acc[t] = vzero;

#pragma unroll 4
  for (int k0 = 0; k0 < D_FEAT; k0 += 4) {
    const int ka = k0 + (half << 1);
    v2f a;
    a.x = xrow[ka] * s;
    a.y = xrow[ka + 1] * s;
    const float* __restrict__ w0 = W + (size_t)ka * D_FEAT + l16;
#pragma unroll
    for (int t = 0; t < 8; ++t) {
      v2f b;
      b.x = w0[t * 16];            // W[ka    ][t*16 + l16]
      b.y = w0[t * 16 + D_FEAT];   // W[ka + 1][t*16 + l16]
      acc[t] = __builtin_amdgcn_wmma_f32_16x16x4_f32(
          false, a, false, b, (short)0, acc[t], false, false);
    }
  }

  const int rbase = row0 + (half << 3);
#pragma unroll
  for (int t = 0; t < 8; ++t) {
#pragma unroll
    for (int j = 0; j < 8; ++j) {
      h[(size_t)(rbase + j) * D_FEAT + t * 16 + l16] = acc[t][j];
    }
  }
}

// -------------------- edge scatter:  out[dst] += h[src] * nd[dst] ----------
// One wave per edge; each lane handles 4 contiguous features (float4 gather,
// 4 native f32 atomics). h_tmp (25.6 MB) and the 25.6 MB dst slice both sit
// in the 192 MB L2, so this phase is L2-atomic bound, not HBM bound.

__global__ __launch_bounds__(256)
void scatter_kernel(const float* __restrict__ h, const int* __restrict__ src,
                    const int* __restrict__ dst, const float* __restrict__ ndv,
                    float* __restrict__ out, int nE) {
  int gid  = blockIdx.x * 256 + threadIdx.x;
  int e    = gid >> 5;
  if (e >= nE) return;
  int lane = gid & 31;
  int s_ = src[e];
  int d_ = dst[e];
  float w = ndv[d_];
  const float4 v = ((const float4*)(h + (size_t)s_ * D_FEAT))[lane];
  float* op = out + (size_t)d_ * D_FEAT + lane * 4;
  atomAddF32(op + 0, v.x * w);
  atomAddF32(op + 1, v.y * w);
  atomAddF32(op + 2, v.z * w);
  atomAddF32(op + 3, v.w * w);
}

// -------------------- BatchNorm (training stats) + PReLU -------------------

__global__ __launch_bounds__(256)
void bn_stats_kernel(const float* __restrict__ x, float* __restrict__ sums,
                     float* __restrict__ sumsq, int rows) {
  const int col = threadIdx.x & (D_FEAT - 1);
  const int rh  = threadIdx.x >> 7;   // two row-lanes per block share a column
  float s = 0.f, q = 0.f;
  for (int r = blockIdx.x * 2 + rh; r < rows; r += gridDim.x * 2) {
    float v = x[(size_t)r * D_FEAT + col];
    s += v;
    q += v * v;
  }
  __shared__ float ls[256];
  __shared__ float lq[256];
  ls[threadIdx.x] = s;
  lq[threadIdx.x] = q;
  __syncthreads();
  if (threadIdx.x < D_FEAT) {
    atomAddF32(&sums[col],  ls[threadIdx.x] + ls[threadIdx.x + D_FEAT]);
    atomAddF32(&sumsq[col], lq[threadIdx.x] + lq[threadIdx.x + D_FEAT]);
  }
}

__global__ __launch_bounds__(128)
void bn_finalize_kernel(const float* __restrict__ sums, const float* __restrict__ sumsq,
                        float* __restrict__ mu, float* __restrict__ inv,
                        float invN, int n) {
  int i = blockIdx.x * 128 + threadIdx.x;
  if (i >= n) return;
  float m   = sums[i] * invN;
  float var = sumsq[i] * invN - m * m;
  mu[i]  = m;
  inv[i] = rsqrtf(var + 1e-5f);
}

__global__ __launch_bounds__(256)
void bn_prelu_kernel(float* __restrict__ x, const float* __restrict__ mu,
                     const float* __restrict__ inv, const float* __restrict__ gamma,
                     const float* __restrict__ beta, const float* __restrict__ alpha,
                     int total) {
  int i = blockIdx.x * 256 + threadIdx.x;
  if (i >= total) return;
  int c = i & (D_FEAT - 1);
  float y = (x[i] - mu[c]) * inv[c] * gamma[c] + beta[c];
  float a = alpha[0];
  x[i] = (y >= 0.f) ? y : a * y;
}

// ---------------------------------------------------------------------------

extern "C" void kernel_launch(void* const* d_in, const int* in_sizes, int n_in,
                              void* d_out, int out_size, void* d_ws, size_t ws_size,
                              hipStream_t stream) {
  (void)n_in; (void)ws_size;
  const int N_ = in_sizes[0] / D_FEAT;   // 50000 nodes per ntype (multiple of 16)

  float* out = (float*)d_out;
  float* ws  = (float*)d_ws;

  // Workspace layout (floats):
  //  [0, 16N)            : per-conv ns/nd arrays (int counts during pass 1)
  //  [16N, 16N + N*128)  : h_tmp  (transformed src features, L2-resident)
  //  then 5*128 sums, 5*128 sumsq, 5*128 mu, 5*128 inv
  float* norms = ws;
  float* h_tmp = ws + 16 * (size_t)N_;
  float* stats = h_tmp + (size_t)N_ * D_FEAT;
  float* sums  = stats;
  float* sumsq = stats + 5 * D_FEAT;
  float* mu    = stats + 10 * D_FEAT;
  float* inv   = stats + 15 * D_FEAT;

  hipMemsetAsync(d_out, 0, (size_t)out_size * sizeof(float), stream);
  hipMemsetAsync(norms, 0, 16 * (size_t)N_ * sizeof(float), stream);
  hipMemsetAsync(stats, 0, 10 * (size_t)D_FEAT * sizeof(float), stream);

  // Input indices (setup_inputs dict order; params dict flattened in order):
  //  0..4  h_{drug,disease,cpm,chp,gene}
  //  5..20 (src,dst) for dd,ddis,disdis,cc,cchp,chpchp,dg,gg
  //  21..36 W_r,b_r pairs; 37..48 gamma/beta/alpha for blocks A,B,C,E
  struct Conv { int x, w, s, d, slice; };
  const Conv convs[8] = {
    {0, 21,  5,  6, 0},  // dd      -> drug    (block A)
    {0, 23,  7,  8, 1},  // ddis    -> disease (block A)
    {1, 25,  9, 10, 1},  // disdis  -> disease (block A)
    {2, 27, 11, 12, 2},  // cc      -> cpm     (block B)
    {2, 29, 13, 14, 3},  // cchp    -> chp     (block C)
    {3, 31, 15, 16, 3},  // chpchp  -> chp     (block C)
    {0, 33, 17, 18, 4},  // dg      -> gene    (block E)
    {4, 35, 19, 20, 4},  // gg      -> gene    (block E)
  };

  // Pass 1: degree counts (int atomics), then convert in place to rsqrt norms.
  for (int c = 0; c < 8; ++c) {
    int nE = in_sizes[convs[c].s];
    int* cs = (int*)(norms + (size_t)(2 * c) * N_);
    int* cd = (int*)(norms + (size_t)(2 * c + 1) * N_);
    count_deg_kernel<<<(nE + 255) / 256, 256, 0, stream>>>(
        (const int*)d_in[convs[c].s], (const int*)d_in[convs[c].d], cs, cd, nE);
  }
  {
    int n = 16 * N_;
    finalize_norm_kernel<<<(n + 255) / 256, 256, 0, stream>>>(norms, n);
  }

  // Pass 2: per relation, WMMA GEMM into h_tmp then edge scatter into d_out.
  // (GraphConv bias is omitted: a per-column constant cancels in BatchNorm.)
  for (int c = 0; c < 8; ++c) {
    const Conv cv = convs[c];
    int nE = in_sizes[cv.s];
    const float* nsv = norms + (size_t)(2 * c) * N_;
    const float* ndv = norms + (size_t)(2 * c + 1) * N_;
    gemm_ns_wmma_kernel<<<N_ / 16, 32, 0, stream>>>(
        (const float*)d_in[cv.x], nsv, (const float*)d_in[cv.w], h_tmp);
    float* slice = out + (size_t)cv.slice * N_ * D_FEAT;
    long threads = (long)nE * 32;
    scatter_kernel<<<(int)((threads + 255) / 256), 256, 0, stream>>>(
        h_tmp, (const int*)d_in[cv.s], (const int*)d_in[cv.d], ndv, slice, nE);
  }

  // Pass 3: BatchNorm (per-slice batch stats) + PReLU, in place on d_out.
  for (int sIdx = 0; sIdx < 5; ++sIdx) {
    const float* slice = out + (size_t)sIdx * N_ * D_FEAT;
    bn_stats_kernel<<<256, 256, 0, stream>>>(
        slice, sums + sIdx * D_FEAT, sumsq + sIdx * D_FEAT, N_);
  }
  bn_finalize_kernel<<<(5 * D_FEAT + 127) / 128, 128, 0, stream>>>(
      sums, sumsq, mu, inv, 1.0f / (float)N_, 5 * D_FEAT);

  const int blk_param[5] = {37, 37, 40, 43, 46};  // gamma idx (beta=+1, alpha=+2)
  for (int sIdx = 0; sIdx < 5; ++sIdx) {
    float* slice = out + (size_t)sIdx * N_ * D_FEAT;
    int total = N_ * D_FEAT;
    bn_prelu_kernel<<<(total + 255) / 256, 256, 0, stream>>>(
        slice, mu + sIdx * D_FEAT, inv + sIdx * D_FEAT,
        (const float*)d_in[blk_param[sIdx]],
        (const float*)d_in[blk_param[sIdx] + 1],
        (const float*)d_in[blk_param[sIdx] + 2], total);
  }
}